// GCNBlock2_10187662426715
// MI455X (gfx1250) — compile-verified
//
#include <hip/hip_runtime.h>

#define DFEAT 128

typedef float v2f __attribute__((ext_vector_type(2)));
typedef float v8f __attribute__((ext_vector_type(8)));

// ---------------------------------------------------------------- utilities
__global__ void zero_f32(float* __restrict__ p, size_t n) {
  size_t i = (size_t)blockIdx.x * blockDim.x + threadIdx.x;
  if (i < n) p[i] = 0.0f;
}

__global__ void degree_kernel(const int* __restrict__ src, const int* __restrict__ dst,
                              float* __restrict__ degS, float* __restrict__ degD, int nE) {
  int e = blockIdx.x * blockDim.x + threadIdx.x;
  if (e < nE) {
    atomicAdd(&degS[src[e]], 1.0f);
    atomicAdd(&degD[dst[e]], 1.0f);
  }
}

__global__ void norm_kernel(float* __restrict__ p, int n) {
  int i = blockIdx.x * blockDim.x + threadIdx.x;
  if (i < n) {
    float v = p[i];
    p[i] = (v > 0.0f) ? rsqrtf(v) : 0.0f;   // deg^-1/2, 0 for isolated nodes
  }
}

// ---------------------------------------------------------------- async copy helpers
// CDNA5 async memory->LDS DMA (no VGPR round trip), tracked by ASYNCcnt.
// GV mode: global_load_async_to_lds_b128 vdst(lds byte offset), vaddr(64b addr), off
__device__ __forceinline__ void async_ld_lds_b128(unsigned int lds_off, const void* gaddr) {
  asm volatile("global_load_async_to_lds_b128 %0, %1, off"
               :
               : "v"(lds_off), "v"(gaddr)
               : "memory");
}
__device__ __forceinline__ void wait_asynccnt0() {
  asm volatile("s_wait_asynccnt 0x0" ::: "memory");
}

// ---------------------------------------------------------------- fp32 WMMA GEMM
// H[nRows x 128] = X[nRows x 128] @ W[128 x 128]
// 4 waves / block, each wave owns a 16-row M tile and all 8 N tiles.
// V_WMMA_F32_16X16X4_F32: A 16x4 f32 (2 VGPR), B 4x16 f32 (2 VGPR), C/D 16x16 f32 (8 VGPR).
__global__ __launch_bounds__(128) void gcn_gemm_wmma_f32(
    const float* __restrict__ X, const float* __restrict__ W,
    float* __restrict__ H, int nRows)
{
  __shared__ float sW[DFEAT * DFEAT];          // 64 KB of the 320 KB/WGP LDS

  // Asynchronously DMA W (row-major 128x128 f32 = 64KB) into LDS:
  // 128 threads x 32 chunks x 16B. Each wave drains its own ASYNCcnt, then barrier.
  {
    const unsigned int sbase = (unsigned int)(size_t)&sW[0];   // LDS aperture: low 32 bits = LDS offset
    for (int i = 0; i < 32; ++i) {
      const int idx = threadIdx.x + i * 128;                   // 16B chunk index
      async_ld_lds_b128(sbase + idx * 16, (const char*)W + (size_t)idx * 16);
    }
    wait_asynccnt0();
  }
  __syncthreads();

  const int wave  = threadIdx.x >> 5;
  const int lane  = threadIdx.x & 31;
  const int m     = lane & 15;                 // M index (A) / N index (B,C)
  const int khalf = (lane >> 4) << 1;          // 0 for lanes 0-15, 2 for 16-31
  const int row0  = blockIdx.x * 64 + wave * 16;
  if (row0 >= nRows) return;                   // wave-uniform (nRows % 16 == 0)

  v8f acc[8];
#pragma unroll
  for (int n = 0; n < 8; ++n) acc[n] = (v8f){0,0,0,0,0,0,0,0};

  const float* xrow = X + (size_t)(row0 + m) * DFEAT;

  for (int k0 = 0; k0 < DFEAT; k0 += 4) {
    // A fragment: lane holds X[row0+m][k0+khalf .. +1]  (8B-aligned -> b64 load)
    const float2 av = *(const float2*)(xrow + k0 + khalf);
    v2f a; a.x = av.x; a.y = av.y;
#pragma unroll
    for (int n = 0; n < 8; ++n) {
      v2f b;                                   // B fragment: W[k0+khalf+v][n*16+m]
      b.x = sW[(k0 + khalf + 0) * DFEAT + n * 16 + m];
      b.y = sW[(k0 + khalf + 1) * DFEAT + n * 16 + m];
      acc[n] = __builtin_amdgcn_wmma_f32_16x16x4_f32(
          /*neg_a=*/false, a, /*neg_b=*/false, b,
          /*c_mod=*/(short)0, acc[n], /*reuse_a=*/false, /*reuse_b=*/false);
    }
  }

  // C/D layout: VGPR v, lanes 0-15 -> M=v, lanes 16-31 -> M=v+8; N = lane%16
  const int rbase = row0 + ((lane >> 4) << 3);
#pragma unroll
  for (int n = 0; n < 8; ++n) {
#pragma unroll
    for (int v = 0; v < 8; ++v) {
      H[(size_t)(rbase + v) * DFEAT + n * 16 + m] = acc[n][v];
    }
  }
}

// ---------------------------------------------------------------- edge phase
// agg[dst[e]] += h[src[e]] * norm_src[src[e]] ; one wave per edge, 4 floats/lane.
__global__ __launch_bounds__(256) void scatter_kernel(
    const float* __restrict__ h, const int* __restrict__ src, const int* __restrict__ dst,
    const float* __restrict__ normS, float* __restrict__ agg, int nE)
{
  size_t t = (size_t)blockIdx.x * blockDim.x + threadIdx.x;
  int e    = (int)(t >> 5);
  int lane = (int)(t & 31);
  if (e >= nE) return;
  int s = src[e];
  int d = dst[e];
  float ns = normS[s];
  const float4 hv = *(const float4*)(h + (size_t)s * DFEAT + lane * 4);
  float* ap = agg + (size_t)d * DFEAT + lane * 4;
  atomicAdd(ap + 0, hv.x * ns);
  atomicAdd(ap + 1, hv.y * ns);
  atomicAdd(ap + 2, hv.z * ns);
  atomicAdd(ap + 3, hv.w * ns);
}

// out = relu(agg * norm_dst + b)
__global__ void finalize_relu_kernel(const float* __restrict__ agg, const float* __restrict__ normD,
                                     const float* __restrict__ bias, float* __restrict__ out, size_t total)
{
  size_t i = (size_t)blockIdx.x * blockDim.x + threadIdx.x;
  if (i >= total) return;
  size_t node = i >> 7;
  int f = (int)(i & (DFEAT - 1));
  float v = agg[i] * normD[node] + bias[f];
  out[i] = fmaxf(v, 0.0f);
}

// inout = inout * norm_dst + b  (in place, final layer)
__global__ void finalize_inplace_kernel(float* __restrict__ inout, const float* __restrict__ normD,
                                        const float* __restrict__ bias, size_t total)
{
  size_t i = (size_t)blockIdx.x * blockDim.x + threadIdx.x;
  if (i >= total) return;
  size_t node = i >> 7;
  int f = (int)(i & (DFEAT - 1));
  inout[i] = inout[i] * normD[node] + bias[f];
}

// ---------------------------------------------------------------- launch
extern "C" void kernel_launch(void* const* d_in, const int* in_sizes, int n_in,
                              void* d_out, int out_size, void* d_ws, size_t ws_size,
                              hipStream_t stream) {
  const float* x  = (const float*)d_in[0];
  const int*   src = (const int*)d_in[1];
  const int*   dst = (const int*)d_in[2];
  const float* W1 = (const float*)d_in[3];
  const float* b1 = (const float*)d_in[4];
  const float* W2 = (const float*)d_in[5];
  const float* b2 = (const float*)d_in[6];
  float* out = (float*)d_out;

  const int nNodes = in_sizes[0] / DFEAT;
  const int nE     = in_sizes[1];
  const size_t nd  = (size_t)nNodes * DFEAT;

  float* ws    = (float*)d_ws;
  float* normS = ws;                       // [nNodes]
  float* normD = ws + nNodes;              // [nNodes]
  float* bufA  = ws + 2 * (size_t)nNodes;  // [nNodes*DFEAT]
  float* bufB  = bufA + nd;                // [nNodes*DFEAT]

  const int B = 256;
  dim3 blk(B);

  // degrees -> norms
  zero_f32<<<(2 * nNodes + B - 1) / B, blk, 0, stream>>>(normS, (size_t)2 * nNodes);
  degree_kernel<<<(nE + B - 1) / B, blk, 0, stream>>>(src, dst, normS, normD, nE);
  norm_kernel<<<(2 * nNodes + B - 1) / B, blk, 0, stream>>>(normS, 2 * nNodes);

  const int gemmGrid = (nNodes + 63) / 64;
  const int edgeGrid = (int)(((size_t)nE * 32 + B - 1) / B);
  const int ndGrid   = (int)((nd + B - 1) / B);

  // ---- layer 1: h1 = relu( Dd^-1/2 A Ds^-1/2 (x W1) + b1 )
  gcn_gemm_wmma_f32<<<gemmGrid, 128, 0, stream>>>(x, W1, bufA, nNodes);
  zero_f32<<<ndGrid, blk, 0, stream>>>(bufB, nd);
  scatter_kernel<<<edgeGrid, blk, 0, stream>>>(bufA, src, dst, normS, bufB, nE);
  finalize_relu_kernel<<<ndGrid, blk, 0, stream>>>(bufB, normD, b1, bufA, nd);

  // ---- layer 2: out = Dd^-1/2 A Ds^-1/2 (h1 W2) + b2
  gcn_gemm_wmma_f32<<<gemmGrid, 128, 0, stream>>>(bufA, W2, bufB, nNodes);
  zero_f32<<<ndGrid, blk, 0, stream>>>(out, nd);
  scatter_kernel<<<edgeGrid, blk, 0, stream>>>(bufB, src, dst, normS, out, nE);
  finalize_inplace_kernel<<<ndGrid, blk, 0, stream>>>(out, normD, b2, nd);
}